// MultiLabelLoss_29978871726288
// MI455X (gfx1250) — compile-verified
//
#include <hip/hip_runtime.h>

// MultiLabelLoss reduction for MI455X (gfx1250, wave32).
// Memory-bound streaming pass: b128 NT loads, per-element math in VALU,
// accumulation through V_WMMA_F32_16X16X4_F32 with B=ones (64-way fused
// reduce-add on the matrix pipe), then hierarchical combine.

typedef __attribute__((ext_vector_type(2))) float v2f;
typedef __attribute__((ext_vector_type(4))) float v4f;
typedef __attribute__((ext_vector_type(8))) float v8f;

__device__ __forceinline__ float fast_ln(float x) {
  // v_log_f32 is log2; scale to natural log.
  return __builtin_amdgcn_logf(x) * 0.69314718055994530942f;
}

struct Quad { float l, pe, ne, ct; };

// t is exactly 0.0 or 1.0, so t*log(p)+(1-t)*log1p(-p) == log(pos ? p : 1-p).
// m is a validity mask (1.0 in-bounds, 0.0 out) so WMMA can run with EXEC all-1s.
__device__ __forceinline__ Quad elem_vals(float p, float t, float m) {
  Quad q;
  const bool pos = (t != 0.0f);
  const float x  = pos ? p : (1.0f - p);
  const float lg = fmaxf(fast_ln(x), -100.0f);   // jnp.clip(log, -100)
  const float d  = p - t;
  const float e  = d * d;
  q.l  = lg * m;
  q.pe = pos ? (e * m) : 0.0f;
  q.ne = pos ? 0.0f     : (e * m);
  q.ct = pos ? m        : 0.0f;
  return q;
}

#define WMMA_RED(C, A)                                                         \
  C = __builtin_amdgcn_wmma_f32_16x16x4_f32(false, (A), false, ones,           \
                                            (short)0, (C), false, false)

__global__ __launch_bounds__(256) void mll_partial(const float* __restrict__ pin,
                                                   const float* __restrict__ tin,
                                                   float* __restrict__ ws,
                                                   int n4) {
  const v4f* __restrict__ p4 = (const v4f*)pin;
  const v4f* __restrict__ t4 = (const v4f*)tin;

  v8f c_bce = {}, c_pos = {}, c_neg = {}, c_cnt = {};
  const v2f ones = {1.0f, 1.0f};

  const int stride = (int)(gridDim.x * blockDim.x);
  const int base   = (int)(blockIdx.x * blockDim.x + threadIdx.x);
  const int iters  = n4 / stride;  // wave-uniform trip count

  for (int i = 0; i < iters; ++i) {
    const int idx = base + i * stride;
    const v4f pv = __builtin_nontemporal_load(p4 + idx);
    const v4f tv = __builtin_nontemporal_load(t4 + idx);

    const Quad q0 = elem_vals(pv.x, tv.x, 1.0f);
    const Quad q1 = elem_vals(pv.y, tv.y, 1.0f);
    const Quad q2 = elem_vals(pv.z, tv.z, 1.0f);
    const Quad q3 = elem_vals(pv.w, tv.w, 1.0f);

    v2f a;
    a.x = q0.l + q1.l;   a.y = q2.l + q3.l;   WMMA_RED(c_bce, a);
    a.x = q0.pe + q1.pe; a.y = q2.pe + q3.pe; WMMA_RED(c_pos, a);
    a.x = q0.ne + q1.ne; a.y = q2.ne + q3.ne; WMMA_RED(c_neg, a);
    a.x = q0.ct + q1.ct; a.y = q2.ct + q3.ct; WMMA_RED(c_cnt, a);
  }

  if (iters * stride < n4) {  // uniform tail; mask by multiply, not EXEC
    const int idx  = base + iters * stride;
    const bool inb = idx < n4;
    const int safe = inb ? idx : 0;
    const v4f pv = __builtin_nontemporal_load(p4 + safe);
    const v4f tv = __builtin_nontemporal_load(t4 + safe);
    const float m = inb ? 1.0f : 0.0f;

    const Quad q0 = elem_vals(pv.x, tv.x, m);
    const Quad q1 = elem_vals(pv.y, tv.y, m);
    const Quad q2 = elem_vals(pv.z, tv.z, m);
    const Quad q3 = elem_vals(pv.w, tv.w, m);

    v2f a;
    a.x = q0.l + q1.l;   a.y = q2.l + q3.l;   WMMA_RED(c_bce, a);
    a.x = q0.pe + q1.pe; a.y = q2.pe + q3.pe; WMMA_RED(c_pos, a);
    a.x = q0.ne + q1.ne; a.y = q2.ne + q3.ne; WMMA_RED(c_neg, a);
    a.x = q0.ct + q1.ct; a.y = q2.ct + q3.ct; WMMA_RED(c_cnt, a);
  }

  // Collapse C tiles. D[m][n] == rowsum(m) replicated over n.
  // Lane l<16 holds D[0..7][l]; lane l>=16 holds D[8..15][l-16].
  float s0 = c_bce[0]+c_bce[1]+c_bce[2]+c_bce[3]+c_bce[4]+c_bce[5]+c_bce[6]+c_bce[7];
  float s1 = c_pos[0]+c_pos[1]+c_pos[2]+c_pos[3]+c_pos[4]+c_pos[5]+c_pos[6]+c_pos[7];
  float s2 = c_neg[0]+c_neg[1]+c_neg[2]+c_neg[3]+c_neg[4]+c_neg[5]+c_neg[6]+c_neg[7];
  float s3 = c_cnt[0]+c_cnt[1]+c_cnt[2]+c_cnt[3]+c_cnt[4]+c_cnt[5]+c_cnt[6]+c_cnt[7];
  s0 += __shfl_xor(s0, 16, 32);
  s1 += __shfl_xor(s1, 16, 32);
  s2 += __shfl_xor(s2, 16, 32);
  s3 += __shfl_xor(s3, 16, 32);

  __shared__ float red[8][4];  // 256 threads -> 8 wave32
  const int wave = (int)(threadIdx.x >> 5);
  const int lane = (int)(threadIdx.x & 31);
  if (lane == 0) {
    red[wave][0] = s0; red[wave][1] = s1; red[wave][2] = s2; red[wave][3] = s3;
  }
  __syncthreads();
  if (threadIdx.x == 0) {
    float b = 0.f, P = 0.f, Nn = 0.f, C = 0.f;
    #pragma unroll
    for (int w = 0; w < 8; ++w) {
      b += red[w][0]; P += red[w][1]; Nn += red[w][2]; C += red[w][3];
    }
    float* slot = ws + (size_t)blockIdx.x * 4;
    slot[0] = b; slot[1] = P; slot[2] = Nn; slot[3] = C;
  }
}

__global__ __launch_bounds__(256) void mll_final(const float* __restrict__ ws,
                                                 float* __restrict__ out,
                                                 int nblocks, float n_total) {
  __shared__ double sb[256], sp[256], sn[256], sc[256];
  double b = 0.0, p = 0.0, nn = 0.0, c = 0.0;
  for (int i = (int)threadIdx.x; i < nblocks; i += 256) {
    const float* slot = ws + (size_t)i * 4;
    b += slot[0]; p += slot[1]; nn += slot[2]; c += slot[3];
  }
  const int tid = (int)threadIdx.x;
  sb[tid] = b; sp[tid] = p; sn[tid] = nn; sc[tid] = c;
  __syncthreads();
  for (int off = 128; off > 0; off >>= 1) {
    if (tid < off) {
      sb[tid] += sb[tid + off];
      sp[tid] += sp[tid + off];
      sn[tid] += sn[tid + off];
      sc[tid] += sc[tid + off];
    }
    __syncthreads();
  }
  if (tid == 0) {
    const double N   = (double)n_total;
    const double bce = -sb[0] / N;                       // mean of -(accumulated logs)
    const double cp  = sc[0];
    const double cn  = N - cp;
    const double fp_pen = sp[0] / fmax(cp, 1.0);         // masked MSE over t!=0
    const double fn_pen = sn[0] / fmax(cn, 1.0);         // masked MSE over t==0
    out[0] = (float)(bce + 0.1 * fn_pen + 0.1 * fp_pen);
  }
}

extern "C" void kernel_launch(void* const* d_in, const int* in_sizes, int n_in,
                              void* d_out, int out_size, void* d_ws, size_t ws_size,
                              hipStream_t stream) {
  const float* p = (const float*)d_in[0];  // "input"
  const float* t = (const float*)d_in[1];  // "target"
  float* out = (float*)d_out;
  float* ws  = (float*)d_ws;

  const int n  = in_sizes[0];      // 4096*8192 = 33,554,432
  const int n4 = n / 4;            // float4 vectors (n divisible by 4)

  int nblocks = 2048;              // 2048 blocks * 16B partials = 32 KB of ws
  const size_t need = (size_t)nblocks * 4 * sizeof(float);
  if (ws_size < need) nblocks = (int)(ws_size / (4 * sizeof(float)));

  mll_partial<<<nblocks, 256, 0, stream>>>(p, t, ws, n4);
  mll_final<<<1, 256, 0, stream>>>(ws, out, nblocks, (float)n);
}